// GATModel_31121333027298
// MI455X (gfx1250) — compile-verified
//
#include <hip/hip_runtime.h>

typedef __attribute__((ext_vector_type(2))) float v2f;
typedef __attribute__((ext_vector_type(8))) float v8f;

#define NN 100000
#define FF 256          // input dim
#define HC 128          // H*C
#define NEG_SLOPE 0.2f
#define ENC_NEG_INF 0x007FFFFFu   // enc(-inf)

// Monotonic float->uint mapping: preserves total order, lets us use the
// single-instruction integer GLOBAL_ATOMIC_MAX_U32 for the segment max.
__device__ __forceinline__ unsigned enc_f32(float x) {
  unsigned u = __float_as_uint(x);
  return u ^ ((u & 0x80000000u) ? 0xFFFFFFFFu : 0x80000000u);
}
__device__ __forceinline__ float dec_f32(unsigned m) {
  unsigned u = (m & 0x80000000u) ? (m ^ 0x80000000u) : ~m;
  return __uint_as_float(u);
}
// Hardware fp32 atomic add (GLOBAL_ATOMIC_ADD_F32), agent scope hits in L2.
__device__ __forceinline__ void atomic_add_f32(float* p, float v) {
  __hip_atomic_fetch_add(p, v, __ATOMIC_RELAXED, __HIP_MEMORY_SCOPE_AGENT);
}

// ---------------------------------------------------------------- init ------
__global__ void __launch_bounds__(256) k_init(float* __restrict__ out,
                                              unsigned* __restrict__ maxbits,
                                              float* __restrict__ denom,
                                              const float* __restrict__ bias) {
  int tid = blockIdx.x * blockDim.x + threadIdx.x;
  if (tid < NN * HC) out[tid] = bias[tid & (HC - 1)];
  if (tid < NN * 2) { maxbits[tid] = ENC_NEG_INF; denom[tid] = 0.0f; }
}

// ------------------------------------------------- fused projection GEMM ----
// proj[n][0:128]   = x[n] @ Wl.T + bl   (xl)
// proj[n][128:256] = x[n] @ Wr.T + br   (xr)
// One block = 16 rows of x; 8 waves x two 16-col tiles = 256 output cols.
__global__ void __launch_bounds__(256) k_proj(const float* __restrict__ x,
                                              const float* __restrict__ Wl,
                                              const float* __restrict__ bl,
                                              const float* __restrict__ Wr,
                                              const float* __restrict__ br,
                                              float* __restrict__ proj) {
  __shared__ __align__(16) float xs[16 * 260];   // stride 260 -> conflict free
  const int m0 = blockIdx.x * 16;
  const int t  = threadIdx.x;

  // Stage the 16x256 x tile into LDS (1024 float4, 4 per thread).
#pragma unroll
  for (int i = 0; i < 4; ++i) {
    int idx = i * 256 + t;
    int row = idx >> 6;
    int c4  = (idx & 63) << 2;
    float4 v = *(const float4*)(x + (long)(m0 + row) * FF + c4);
    *(float4*)(&xs[row * 260 + c4]) = v;
  }
  __syncthreads();

  const int lane = t & 31;
  const int wave = t >> 5;
  const int r    = lane & 15;          // M (A) / N (B,C,D) index
  const int koff = (lane >> 4) << 1;   // lanes 16-31 hold K+2,K+3

  for (int t2 = 0; t2 < 2; ++t2) {
    const int n0  = wave * 32 + t2 * 16;
    const int col = n0 + r;
    const float* Wp;
    float bv;
    if (col < HC) { Wp = Wl + (long)col * FF;        bv = bl[col]; }
    else          { Wp = Wr + (long)(col - HC) * FF; bv = br[col - HC]; }
    __builtin_prefetch(Wp, 0, 0);      // global_prefetch_b8 into L2

    v8f acc;
#pragma unroll
    for (int v = 0; v < 8; ++v) acc[v] = bv;   // bias: per-column broadcast

#pragma unroll 8
    for (int k0 = 0; k0 < FF; k0 += 4) {
      v2f a, b;
      a[0] = xs[r * 260 + k0 + koff];
      a[1] = xs[r * 260 + k0 + koff + 1];
      b[0] = Wp[k0 + koff];
      b[1] = Wp[k0 + koff + 1];
      acc = __builtin_amdgcn_wmma_f32_16x16x4_f32(
          false, a, false, b, (short)0, acc, false, false);
    }

    const int rowbase = m0 + ((lane >> 4) << 3);   // M = v + 8*(lane>=16)
#pragma unroll
    for (int v = 0; v < 8; ++v)
      proj[(long)(rowbase + v) * FF + col] = acc[v];
  }
}

// --------------------------------------- per-edge logits + segment max ------
// One wave per edge. Lanes 0-15: head 0 (c = lane*4..), lanes 16-31: head 1.
__global__ void __launch_bounds__(256) k_logits(const int* __restrict__ ei,
                                                const float* __restrict__ proj,
                                                const float* __restrict__ att,
                                                float* __restrict__ logits,
                                                unsigned* __restrict__ maxbits,
                                                int Eraw, int Etot) {
  int wid = blockIdx.x * 8 + (threadIdx.x >> 5);
  if (wid >= Etot) return;
  int lane = threadIdx.x & 31;

  int src, dst;
  if (wid < Eraw) { src = ei[wid]; dst = ei[Eraw + wid]; }
  else            { src = dst = wid - Eraw; }   // self loop

  int j0 = lane << 2;
  float4 a = *(const float4*)(proj + (long)src * FF + j0);        // xl[src]
  float4 b = *(const float4*)(proj + (long)dst * FF + HC + j0);   // xr[dst]
  float4 w = *(const float4*)(att + j0);

  float s = 0.0f, e;
  e = a.x + b.x; e = e > 0.0f ? e : NEG_SLOPE * e; s += w.x * e;
  e = a.y + b.y; e = e > 0.0f ? e : NEG_SLOPE * e; s += w.y * e;
  e = a.z + b.z; e = e > 0.0f ? e : NEG_SLOPE * e; s += w.z * e;
  e = a.w + b.w; e = e > 0.0f ? e : NEG_SLOPE * e; s += w.w * e;

  // reduce within each 16-lane head group (wave32 shuffles)
  for (int m = 1; m < 16; m <<= 1) s += __shfl_xor(s, m, 32);

  if ((lane & 15) == 0) {
    int h = lane >> 4;
    logits[wid * 2 + h] = s;
    atomicMax(&maxbits[dst * 2 + h], enc_f32(s));
  }
}

// ---------------------------------- exp(logit - max) + segment sum ----------
__global__ void __launch_bounds__(256) k_exp(const int* __restrict__ ei,
                                             const unsigned* __restrict__ maxbits,
                                             float* __restrict__ lp,
                                             float* __restrict__ denom,
                                             int Eraw, int Etot) {
  int tid = blockIdx.x * blockDim.x + threadIdx.x;
  if (tid >= 2 * Etot) return;
  int e = tid >> 1, h = tid & 1;
  int dst = (e < Eraw) ? ei[Eraw + e] : (e - Eraw);
  float m = dec_f32(maxbits[dst * 2 + h]);
  float p = __expf(lp[tid] - m);
  lp[tid] = p;                       // overwrite logits with unnorm. weight
  atomic_add_f32(&denom[dst * 2 + h], p);
}

// --------------------------- weighted scatter-add aggregation ---------------
// One wave per edge, 4 channels per lane; proj is L2-resident (102MB < 192MB).
__global__ void __launch_bounds__(256) k_agg(const int* __restrict__ ei,
                                             const float* __restrict__ proj,
                                             const float* __restrict__ p,
                                             const float* __restrict__ denom,
                                             float* __restrict__ out,
                                             int Eraw, int Etot) {
  int wid = blockIdx.x * 8 + (threadIdx.x >> 5);
  if (wid >= Etot) return;
  int lane = threadIdx.x & 31;

  int src, dst;
  if (wid < Eraw) { src = ei[wid]; dst = ei[Eraw + wid]; }
  else            { src = dst = wid - Eraw; }

  int j0 = lane << 2;
  int h  = lane >> 4;
  float alpha = p[wid * 2 + h] / denom[dst * 2 + h];
  float4 xl = *(const float4*)(proj + (long)src * FF + j0);
  float* o = out + (long)dst * HC + j0;
  atomic_add_f32(o + 0, alpha * xl.x);
  atomic_add_f32(o + 1, alpha * xl.y);
  atomic_add_f32(o + 2, alpha * xl.z);
  atomic_add_f32(o + 3, alpha * xl.w);
}

// ---------------------------------------------------------------------------
extern "C" void kernel_launch(void* const* d_in, const int* in_sizes, int n_in,
                              void* d_out, int out_size, void* d_ws, size_t ws_size,
                              hipStream_t stream) {
  const float* x    = (const float*)d_in[0];
  const float* Wl   = (const float*)d_in[1];
  const float* bl   = (const float*)d_in[2];
  const float* Wr   = (const float*)d_in[3];
  const float* br   = (const float*)d_in[4];
  const float* att  = (const float*)d_in[5];
  const float* bias = (const float*)d_in[6];
  const int*   ei   = (const int*)d_in[7];
  float* out = (float*)d_out;

  const int Eraw = in_sizes[7] / 2;
  const int Etot = Eraw + NN;

  // workspace layout: proj[N*256] | p/logits[Etot*2] | maxbits[N*2] | denom[N*2]
  float*    proj    = (float*)d_ws;
  float*    logits  = proj + (size_t)NN * FF;
  unsigned* maxbits = (unsigned*)(logits + (size_t)Etot * 2);
  float*    denom   = (float*)(maxbits + (size_t)NN * 2);

  k_init  <<<(NN * HC + 255) / 256, 256, 0, stream>>>(out, maxbits, denom, bias);
  k_proj  <<<NN / 16,               256, 0, stream>>>(x, Wl, bl, Wr, br, proj);
  k_logits<<<(Etot + 7) / 8,        256, 0, stream>>>(ei, proj, att, logits, maxbits, Eraw, Etot);
  k_exp   <<<(2 * Etot + 255) / 256,256, 0, stream>>>(ei, maxbits, logits, denom, Eraw, Etot);
  k_agg   <<<(Etot + 7) / 8,        256, 0, stream>>>(ei, proj, logits, denom, out, Eraw, Etot);
}